// GraphSageConv_26663156973805
// MI455X (gfx1250) — compile-verified
//
#include <hip/hip_runtime.h>

typedef __attribute__((ext_vector_type(2))) float v2f;
typedef __attribute__((ext_vector_type(4))) float v4f;
typedef __attribute__((ext_vector_type(8))) float v8f;

#define N_NODES 100000
#define E_EDGES 1600000
#define IN_F    256
#define OUT_F   128
#define ROWS_PER_BLOCK 32          // 100000 / 32 = 3125 blocks exactly
#define LDS_STRIDE 260             // 260 % 64 == 4 -> conflict-free column reads

// ---------------------------------------------------------------------------
// GEMM: support = X @ W ; out = support + bias
// grid.x = 3125 ; block = 256 (8 waves, wave w -> cols [16w,16w+16))
// Each wave carries TWO 16x16 accumulators (row tiles rbase / rbase+16) so
// every B-fragment load pair feeds two v_wmma_f32_16x16x4_f32 ops:
// 1 VMEM b32 load per WMMA instead of 2, and W traffic halves.
// ---------------------------------------------------------------------------
__global__ __launch_bounds__(256) void gcn_gemm_bias_kernel(
    const float* __restrict__ X, const float* __restrict__ W,
    const float* __restrict__ bias, float* __restrict__ support,
    float* __restrict__ out)
{
    __shared__ float xs[ROWS_PER_BLOCK * LDS_STRIDE];

    const int rbase = blockIdx.x * ROWS_PER_BLOCK;
    const int tid   = threadIdx.x;

    // Cooperative load of the 32x256 X tile: 2048 float4, 8 per thread.
    {
        const v4f* src = (const v4f*)(X + rbase * IN_F);
        #pragma unroll
        for (int i = 0; i < 8; ++i) {
            int idx = tid + i * 256;        // float4 index 0..2047
            int r   = idx >> 6;             // 64 float4 per row
            int c   = idx & 63;
            v4f v = src[idx];
            *(v4f*)&xs[r * LDS_STRIDE + (c << 2)] = v;
        }
    }
    __syncthreads();

    const int lane = tid & 31;
    const int wave = tid >> 5;
    const int hi   = lane >> 4;          // 0: K=0,1 half ; 1: K=2,3 half
    const int m    = lane & 15;
    const int n    = (wave << 4) + m;    // output column

    v8f acc0 = {};                       // rows rbase      .. rbase+15
    v8f acc1 = {};                       // rows rbase + 16 .. rbase+31
    const float* xrow0 = &xs[m * LDS_STRIDE + (hi << 1)];
    const float* xrow1 = &xs[(m + 16) * LDS_STRIDE + (hi << 1)];
    const float* wcol  = W + n + (hi << 1) * OUT_F;

    // A 16x4 layout: lanes0-15 VGPR{0,1}=A[m][k,k+1]; lanes16-31 =A[m][k+2,k+3]
    // B 4x16 layout: lanes0-15 VGPR{0,1}=B[k,k+1][n]; lanes16-31 =B[k+2,k+3][n]
    #pragma unroll 4
    for (int k = 0; k < IN_F; k += 4) {
        v2f b;
        b.x = wcol[k * OUT_F];
        b.y = wcol[(k + 1) * OUT_F];
        v2f a0 = *(const v2f*)(xrow0 + k);
        v2f a1 = *(const v2f*)(xrow1 + k);
        acc0 = __builtin_amdgcn_wmma_f32_16x16x4_f32(
            false, a0, false, b, (short)0, acc0, false, false);
        acc1 = __builtin_amdgcn_wmma_f32_16x16x4_f32(
            false, a1, false, b, (short)0, acc1, false, false);
    }

    // D layout: VGPR v -> row = tile_base + v + 8*hi, col = n
    const float bv = bias[n];
    const int   r0 = rbase + (hi << 3);
    #pragma unroll
    for (int vi = 0; vi < 8; ++vi) {
        float s0   = acc0[vi];
        float s1   = acc1[vi];
        int   off0 = (r0 + vi) * OUT_F + n;
        int   off1 = (r0 + 16 + vi) * OUT_F + n;
        support[off0] = s0;          // pure X@W, feeds the scatter phase
        support[off1] = s1;
        out[off0]     = s0 + bv;     // self term + bias (scatter adds on top)
        out[off1]     = s1 + bv;
    }
}

// ---------------------------------------------------------------------------
// Scatter: out[col[e]] += support[row[e]]  (one wave per edge)
// grid.x = E/8 = 200000 ; block = 256 (8 waves). Lane covers 4 floats.
// support is L2-resident (51.2 MB < 192 MB L2); atomics resolve in L2.
// ---------------------------------------------------------------------------
__global__ __launch_bounds__(256) void gcn_scatter_kernel(
    const float* __restrict__ support, const int* __restrict__ edges,
    float* __restrict__ out)
{
    const int e    = blockIdx.x * 8 + (threadIdx.x >> 5);
    const int lane = threadIdx.x & 31;

    const int row = edges[e];             // edge_index[0][e]
    const int col = edges[E_EDGES + e];   // edge_index[1][e]

    const v4f v = *(const v4f*)(support + row * OUT_F + (lane << 2));
    float* dst  = out + col * OUT_F + (lane << 2);

    atomicAdd(dst + 0, v.x);
    atomicAdd(dst + 1, v.y);
    atomicAdd(dst + 2, v.z);
    atomicAdd(dst + 3, v.w);
}

// ---------------------------------------------------------------------------
extern "C" void kernel_launch(void* const* d_in, const int* in_sizes, int n_in,
                              void* d_out, int out_size, void* d_ws, size_t ws_size,
                              hipStream_t stream)
{
    const float* X     = (const float*)d_in[0];
    const float* W     = (const float*)d_in[1];
    const float* bias  = (const float*)d_in[2];
    const int*   edges = (const int*)d_in[3];
    float*       out   = (float*)d_out;
    float*       supp  = (float*)d_ws;    // N_NODES*OUT_F floats = 51.2 MB

    gcn_gemm_bias_kernel<<<N_NODES / ROWS_PER_BLOCK, 256, 0, stream>>>(
        X, W, bias, supp, out);
    gcn_scatter_kernel<<<E_EDGES / 8, 256, 0, stream>>>(supp, edges, out);
}